// InterpolativeUpsamplerCUDA_77257871720547
// MI455X (gfx1250) — compile-verified
//
#include <hip/hip_runtime.h>
#include <hip/hip_bf16.h>
#include <stdint.h>

#define IN_H  128
#define IN_W  128
#define OUT_W 256

typedef __attribute__((ext_vector_type(4))) float v4f;

// CDNA5 async global->LDS copy (ASYNCcnt-tracked), 16 bytes per lane.
// VDST operand is the LDS byte offset within the group segment; the low 32
// bits of the generic (flat) address of a __shared__ object are exactly that
// offset on AMDGPU (flat LDS addr = {shared_aperture_hi, lds_offset}).
__device__ __forceinline__ void async_copy_b128(uint32_t lds_byte_off, const void* gaddr) {
    asm volatile("global_load_async_to_lds_b128 %0, %1, off"
                 :: "v"(lds_byte_off), "v"(gaddr)
                 : "memory");
}

__device__ __forceinline__ void wait_async_all() {
    asm volatile("s_wait_asynccnt 0" ::: "memory");
}

// Separable 2x smooth upsampler:
//   even phase weights (offsets -1,0):    [1,1]/2
//   odd  phase weights (offsets -1,0,+1): [1,6,1]/8
// applied per axis, with reflect-mirroring at borders.
__global__ __launch_bounds__(256) void upsample2x_smooth_kernel(
        const float* __restrict__ x, float* __restrict__ y) {
    __shared__ float tile[IN_H * IN_W];   // 64 KiB: one full input plane

    const int plane = blockIdx.x;
    const float* __restrict__ src = x + (size_t)plane * (IN_H * IN_W);
    float* __restrict__ dst = y + (size_t)plane * (size_t)(2 * IN_H) * OUT_W;

    const int tid = threadIdx.x;
    const uint32_t lds_base = (uint32_t)(uintptr_t)(&tile[0]);

    // Stage the whole 128x128 fp32 plane into LDS: 4096 x b128 transfers,
    // 16 per thread, all tracked on ASYNCcnt.
    #pragma unroll
    for (int i = 0; i < 16; ++i) {
        const uint32_t boff = (uint32_t)(tid + 256 * i) * 16u;  // bytes
        async_copy_b128(lds_base + boff, (const char*)src + boff);
    }
    wait_async_all();   // this wave's async copies landed in LDS
    __syncthreads();    // ... and everyone else's too

    const int tx = tid & 63;   // horizontal group: input cols n0, n0+1
    const int tm = tid >> 6;   // 0..3: row slice within a 4-row strip
    const int n0 = 2 * tx;

    // mirrored column indices (reflect, no edge repeat)
    const int cm1 = (n0 == 0) ? 1 : n0 - 1;                       // n0-1
    const int cp2 = (n0 == IN_W - 2) ? (IN_W - 2) : n0 + 2;       // n0+2

    for (int mb = 0; mb < IN_H; mb += 4) {
        const int m   = mb + tm;
        const int rm1 = (m == 0) ? 1 : m - 1;                     // mirror
        const int rp1 = (m == IN_H - 1) ? (IN_H - 2) : m + 1;     // mirror

        // horizontal pass on rows m-1, m, m+1 for 4 output columns
        float hE[3][2], hO[3][2];
        const int rows[3] = { rm1, m, rp1 };
        #pragma unroll
        for (int j = 0; j < 3; ++j) {
            const float* r = &tile[rows[j] * IN_W];
            const float a = r[cm1];
            const float b = r[n0];
            const float c = r[n0 + 1];
            const float d = r[cp2];
            hE[j][0] = (a + b) * 0.5f;                  // out col 2*n0
            hO[j][0] = (a + 6.0f * b + c) * 0.125f;     // out col 2*n0+1
            hE[j][1] = (b + c) * 0.5f;                  // out col 2*n0+2
            hO[j][1] = (b + 6.0f * c + d) * 0.125f;     // out col 2*n0+3
        }

        // vertical pass -> output rows 2m (even phase) and 2m+1 (odd phase)
        v4f rowE, rowO;
        rowE.x = (hE[0][0] + hE[1][0]) * 0.5f;
        rowE.y = (hO[0][0] + hO[1][0]) * 0.5f;
        rowE.z = (hE[0][1] + hE[1][1]) * 0.5f;
        rowE.w = (hO[0][1] + hO[1][1]) * 0.5f;
        rowO.x = (hE[0][0] + 6.0f * hE[1][0] + hE[2][0]) * 0.125f;
        rowO.y = (hO[0][0] + 6.0f * hO[1][0] + hO[2][0]) * 0.125f;
        rowO.z = (hE[0][1] + 6.0f * hE[1][1] + hE[2][1]) * 0.125f;
        rowO.w = (hO[0][1] + 6.0f * hO[1][1] + hO[2][1]) * 0.125f;

        // streaming (non-temporal) coalesced b128 stores: output is write-once
        v4f* pE = (v4f*)(dst + (size_t)(2 * m)     * OUT_W + 4 * tx);
        v4f* pO = (v4f*)(dst + (size_t)(2 * m + 1) * OUT_W + 4 * tx);
        __builtin_nontemporal_store(rowE, pE);
        __builtin_nontemporal_store(rowO, pO);
    }
}

extern "C" void kernel_launch(void* const* d_in, const int* in_sizes, int n_in,
                              void* d_out, int out_size, void* d_ws, size_t ws_size,
                              hipStream_t stream) {
    const float* x = (const float*)d_in[0];   // (16,128,128,128) fp32
    float* y = (float*)d_out;                 // (16,128,256,256) fp32
    const int planes = in_sizes[0] / (IN_H * IN_W);  // 16*128 = 2048
    hipLaunchKernelGGL(upsample2x_smooth_kernel, dim3(planes), dim3(256), 0, stream, x, y);
}